// MemoryModule_26190710571442
// MI455X (gfx1250) — compile-verified
//
#include <hip/hip_runtime.h>
#include <math.h>

// ---------------------------------------------------------------------------
// MI455X (gfx1250) implementation.
//   - All 5 GEMMs on V_WMMA_F32_16X16X4_F32 (full fp32 numerics, matrix pipe).
//   - Per-wave fixed N-stripe: B fragment loaded once per K-step, reused for
//     all M-tiles (4x less weight traffic than per-tile K-loops).
//   - LDS tile staging via GLOBAL_LOAD_ASYNC_TO_LDS_B128 (ASYNCcnt path) --
//     builtin signatures confirmed by round-2 diagnostics:
//       b128: (int4 AS1*, int4 AS3*, imm offset, imm cpol)
//       b32 : (int  AS1*, int  AS3*, imm offset, imm cpol)
//   - Workspace (floats): [0,NC) q/qa | [NC,2NC) xl | [2NC,3NC) a_pre |
//     [3NC,4NC) b_ | kv aliased over [NC,3NC) after xl/a_pre die |
//     [4NC,...) o (B*C*49). Total ~269 MB.
// ---------------------------------------------------------------------------

typedef __attribute__((ext_vector_type(2))) float v2f;
typedef __attribute__((ext_vector_type(8))) float v8f;
typedef int gi4 __attribute__((vector_size(16)));   // GCC-style int4

#define B_   8
#define H_   128
#define W_   128
#define C_   128
#define HW_  (H_ * W_)
#define NPIX (B_ * HW_)
#define NH_  8
#define DH_  16
#define EPS_ 1e-6f

#define V8F_ZERO {0.f,0.f,0.f,0.f,0.f,0.f,0.f,0.f}

// ---- async global->LDS staging (gfx1250 ASYNCcnt path) --------------------
#if defined(__gfx1250__) && \
    __has_builtin(__builtin_amdgcn_global_load_async_to_lds_b128) && \
    __has_builtin(__builtin_amdgcn_global_load_async_to_lds_b32)
#define ASYNC_CP 1
#else
#define ASYNC_CP 0
#endif

__device__ __forceinline__ void cp_b128(const float* g, float* l) {
#if ASYNC_CP
  __builtin_amdgcn_global_load_async_to_lds_b128(
      (__attribute__((address_space(1))) gi4*)(g),
      (__attribute__((address_space(3))) gi4*)(l), 0, 0);
#else
  const float4 v = *(const float4*)g;
  l[0] = v.x; l[1] = v.y; l[2] = v.z; l[3] = v.w;
#endif
}

__device__ __forceinline__ void cp_b32(const float* g, float* l) {
#if ASYNC_CP
  __builtin_amdgcn_global_load_async_to_lds_b32(
      (__attribute__((address_space(1))) int*)(g),
      (__attribute__((address_space(3))) int*)(l), 0, 0);
#else
  *l = *g;
#endif
}

__device__ __forceinline__ void cp_wait() {
#if ASYNC_CP
#if __has_builtin(__builtin_amdgcn_s_wait_asynccnt)
  __builtin_amdgcn_s_wait_asynccnt(0);
#else
  asm volatile("s_wait_asynccnt 0x0" ::: "memory");
#endif
#endif
}

// ---- WMMA fp32 column-strip: one 16-col B stripe vs MT 16-row A tiles -----
// A: row-major [MT*16][lda] LDS tile.  B: row-major [K][ldb] global weights,
// pre-offset to the stripe's first column.
// ISA layout: A 16x4 -> lanes0-15 = M rows, VGPR0/1 = K+2*hi, K+2*hi+1.
//             B 4x16 -> lanes0-15 = N cols, same K striping.
//             C/D    -> VGPR r: lanes0-15 M=r, lanes16-31 M=r+8.
template <int MT>
__device__ __forceinline__ void wmma_strip(const float* At, int lda,
                                           const float* Bm, int ldb, int K,
                                           int lane, v8f acc[MT]) {
  const int half = lane & 15;
  const int hi   = lane >> 4;
  for (int k = 0; k < K; k += 4) {
    v2f b;
    b.x = Bm[(k + 2 * hi) * ldb + half];
    b.y = Bm[(k + 2 * hi + 1) * ldb + half];
#pragma unroll
    for (int mt = 0; mt < MT; ++mt) {
      v2f a;
      a.x = At[(mt * 16 + half) * lda + k + 2 * hi];
      a.y = At[(mt * 16 + half) * lda + k + 2 * hi + 1];
      acc[mt] = __builtin_amdgcn_wmma_f32_16x16x4_f32(
          false, a, false, b, (short)0, acc[mt], false, false);
    }
  }
}

// ---------------------------------------------------------------------------
// Kernel 1: LayerNorm + q = xn@Wq+bq, xl = gelu(xn@Wl+bl). 64 rows/block.
// ---------------------------------------------------------------------------
__global__ __launch_bounds__(256) void k_ln_q_l(
    const float* __restrict__ x, const float* __restrict__ ln_w,
    const float* __restrict__ ln_b,
    const float* __restrict__ Wq, const float* __restrict__ bq,
    const float* __restrict__ Wl, const float* __restrict__ bl,
    float* __restrict__ qo, float* __restrict__ xlo) {
  __shared__ float xt[64 * 132];          // padded stride vs 64 banks
  const int tid = threadIdx.x;
  const long row0 = (long)blockIdx.x * 64;
  __builtin_prefetch(Wq, 0, 3);
  __builtin_prefetch(Wl, 0, 3);
  // stage 64x128 tile
  for (int i = 0; i < 8; ++i) {
    int idx = i * 256 + tid;              // 0..2047
    int r = idx >> 5, c4 = (idx & 31) << 2;
    cp_b128(x + (row0 + r) * C_ + c4, &xt[r * 132 + c4]);
  }
  cp_wait();
  __syncthreads();
  // LayerNorm per row (cheap vs GEMM; 64 rows, one thread each)
  if (tid < 64) {
    float* r = &xt[tid * 132];
    float s = 0.f;
    for (int j = 0; j < C_; ++j) s += r[j];
    const float mu = s * (1.f / C_);
    float v = 0.f;
    for (int j = 0; j < C_; ++j) { float d = r[j] - mu; v += d * d; }
    const float inv = rsqrtf(v * (1.f / C_) + EPS_);
    for (int j = 0; j < C_; ++j)
      r[j] = (r[j] - mu) * inv * ln_w[j] + ln_b[j];
  }
  __syncthreads();
  const int lane = tid & 31, wave = tid >> 5;
  const int half = lane & 15, hi = lane >> 4;
  const int nt = wave;                    // each wave owns one 16-col stripe
  for (int mat = 0; mat < 2; ++mat) {
    const float* Wm = mat ? Wl : Wq;
    const float* bm = mat ? bl : bq;
    v8f acc[4] = {V8F_ZERO, V8F_ZERO, V8F_ZERO, V8F_ZERO};
    wmma_strip<4>(xt, 132, Wm + nt * 16, C_, C_, lane, acc);
    const int col = nt * 16 + half;
    const float bias = bm[col];
#pragma unroll
    for (int mt = 0; mt < 4; ++mt) {
#pragma unroll
      for (int r8 = 0; r8 < 8; ++r8) {
        const long rowg = row0 + mt * 16 + r8 + 8 * hi;
        const float val = acc[mt][r8] + bias;
        if (mat) {  // exact GELU
          xlo[rowg * C_ + col] =
              0.5f * val * (1.f + erff(val * 0.70710678118f));
        } else {
          qo[rowg * C_ + col] = val;
        }
      }
    }
  }
}

// ---------------------------------------------------------------------------
// Kernel 2: dual depthwise 7x7 (pad 3) over xl -> a_pre, b_. NHWC, 4 ch/thread.
// ---------------------------------------------------------------------------
__global__ __launch_bounds__(256) void k_dwconv2(
    const float* __restrict__ xl,
    const float* __restrict__ wA, const float* __restrict__ bA,
    const float* __restrict__ wB, const float* __restrict__ bB,
    float* __restrict__ outA, float* __restrict__ outB) {
  __shared__ float wsA[C_ * 49], wsB[C_ * 49];   // 50 KB LDS
  const int tid = threadIdx.x;
  for (int i = tid; i < C_ * 49; i += 256) {
    cp_b32(wA + i, &wsA[i]);
    cp_b32(wB + i, &wsB[i]);
  }
  cp_wait();
  __syncthreads();
  const long gid = (long)blockIdx.x * 256 + tid;
  const int  c4  = (int)(gid & 31) << 2;
  const long pix = gid >> 5;
  const int  b   = (int)(pix >> 14);
  const int  rem = (int)(pix & 16383);
  const int  y   = rem >> 7, x = rem & 127;
  float a0 = 0, a1 = 0, a2 = 0, a3 = 0;
  float o0 = 0, o1 = 0, o2 = 0, o3 = 0;
  for (int dy = 0; dy < 7; ++dy) {
    const int yy = y + dy - 3;
    if ((unsigned)yy >= 128u) continue;
    for (int dx = 0; dx < 7; ++dx) {
      const int xx = x + dx - 3;
      if ((unsigned)xx >= 128u) continue;
      const float4 v =
          *(const float4*)(xl + ((long)b * HW_ + yy * 128 + xx) * C_ + c4);
      const int t = dy * 7 + dx;
      a0 += v.x * wsA[(c4 + 0) * 49 + t];
      a1 += v.y * wsA[(c4 + 1) * 49 + t];
      a2 += v.z * wsA[(c4 + 2) * 49 + t];
      a3 += v.w * wsA[(c4 + 3) * 49 + t];
      o0 += v.x * wsB[(c4 + 0) * 49 + t];
      o1 += v.y * wsB[(c4 + 1) * 49 + t];
      o2 += v.z * wsB[(c4 + 2) * 49 + t];
      o3 += v.w * wsB[(c4 + 3) * 49 + t];
    }
  }
  float4 ra = {a0 + bA[c4], a1 + bA[c4 + 1], a2 + bA[c4 + 2], a3 + bA[c4 + 3]};
  float4 rb = {o0 + bB[c4], o1 + bB[c4 + 1], o2 + bB[c4 + 2], o3 + bB[c4 + 3]};
  *(float4*)(outA + pix * C_ + c4) = ra;
  *(float4*)(outB + pix * C_ + c4) = rb;
}

// ---------------------------------------------------------------------------
// Kernel 3: a = a_pre@Wa+ba, then qa = q*a written in place over q buffer.
// ---------------------------------------------------------------------------
__global__ __launch_bounds__(256) void k_gemm_qa(
    const float* __restrict__ apre, const float* __restrict__ Wa,
    const float* __restrict__ ba, float* __restrict__ qbuf) {
  __shared__ float at[64 * 132];
  const int tid = threadIdx.x;
  const long row0 = (long)blockIdx.x * 64;
  __builtin_prefetch(Wa, 0, 3);
  for (int i = 0; i < 8; ++i) {
    int idx = i * 256 + tid;
    int r = idx >> 5, c4 = (idx & 31) << 2;
    cp_b128(apre + (row0 + r) * C_ + c4, &at[r * 132 + c4]);
  }
  cp_wait();
  __syncthreads();
  const int lane = tid & 31, wave = tid >> 5;
  const int half = lane & 15, hi = lane >> 4;
  const int nt = wave;
  v8f acc[4] = {V8F_ZERO, V8F_ZERO, V8F_ZERO, V8F_ZERO};
  wmma_strip<4>(at, 132, Wa + nt * 16, C_, C_, lane, acc);
  const int col = nt * 16 + half;
  const float bias = ba[col];
#pragma unroll
  for (int mt = 0; mt < 4; ++mt) {
#pragma unroll
    for (int r8 = 0; r8 < 8; ++r8) {
      const long rowg = row0 + mt * 16 + r8 + 8 * hi;
      const float a = acc[mt][r8] + bias;
      qbuf[rowg * C_ + col] = qbuf[rowg * C_ + col] * a;   // qa = q .* a
    }
  }
}

// ---------------------------------------------------------------------------
// Kernel 4: kv = b_@Wkv+bkv, [NPIX,256]. Written over dead xl/a_pre region.
// ---------------------------------------------------------------------------
__global__ __launch_bounds__(256) void k_gemm_kv(
    const float* __restrict__ bpre, const float* __restrict__ Wkv,
    const float* __restrict__ bkv, float* __restrict__ kv) {
  __shared__ float bt[64 * 132];
  const int tid = threadIdx.x;
  const long row0 = (long)blockIdx.x * 64;
  __builtin_prefetch(Wkv, 0, 3);
  for (int i = 0; i < 8; ++i) {
    int idx = i * 256 + tid;
    int r = idx >> 5, c4 = (idx & 31) << 2;
    cp_b128(bpre + (row0 + r) * C_ + c4, &bt[r * 132 + c4]);
  }
  cp_wait();
  __syncthreads();
  const int lane = tid & 31, wave = tid >> 5;
  const int half = lane & 15, hi = lane >> 4;
  for (int s = 0; s < 2; ++s) {
    const int nt = wave + 8 * s;          // 16 N-stripes across 8 waves
    v8f acc[4] = {V8F_ZERO, V8F_ZERO, V8F_ZERO, V8F_ZERO};
    wmma_strip<4>(bt, 132, Wkv + nt * 16, 2 * C_, C_, lane, acc);
    const int col = nt * 16 + half;
    const float bias = bkv[col];
#pragma unroll
    for (int mt = 0; mt < 4; ++mt) {
#pragma unroll
      for (int r8 = 0; r8 < 8; ++r8) {
        const long rowg = row0 + mt * 16 + r8 + 8 * hi;
        kv[rowg * (2 * C_) + col] = acc[mt][r8] + bias;
      }
    }
  }
}

// ---------------------------------------------------------------------------
// Kernel 5: attention. Block per (b,h); wave per query; lanes stream keys
// with online softmax; shuffle-merge (max,sum,acc[16]). o: [B,C,7,7].
// ---------------------------------------------------------------------------
__global__ __launch_bounds__(256) void k_attn(
    const float* __restrict__ kv, const float* __restrict__ m,
    float* __restrict__ o) {
  __shared__ float mq[49 * DH_];
  const int tid = threadIdx.x;
  const int bh = blockIdx.x, b = bh >> 3, h = bh & 7;
  for (int i = tid; i < 49 * DH_; i += 256) {
    const int q = i >> 4, dd = i & 15;
    mq[i] = m[q * C_ + h * DH_ + dd] * 0.25f;   // * d^-0.5, d=16
  }
  __syncthreads();
  const int lane = tid & 31, wave = tid >> 5;
  for (int q = wave; q < 49; q += 8) {
    float mr[DH_];
#pragma unroll
    for (int dd = 0; dd < DH_; ++dd) mr[dd] = mq[q * DH_ + dd];
    float mmax = -3.0e38f, ssum = 0.f;
    float acc[DH_];
#pragma unroll
    for (int dd = 0; dd < DH_; ++dd) acc[dd] = 0.f;
    for (int p = lane; p < HW_; p += 32) {
      const float* kr = kv + ((long)b * HW_ + p) * (2 * C_) + h * DH_;
      const float* vr = kr + C_;
      float s = 0.f;
#pragma unroll
      for (int dd = 0; dd < DH_; ++dd) s += mr[dd] * kr[dd];
      const float mn = fmaxf(mmax, s);
      const float cs = __expf(mmax - mn);   // rescale old state
      const float ps = __expf(s - mn);
      ssum = ssum * cs + ps;
#pragma unroll
      for (int dd = 0; dd < DH_; ++dd) acc[dd] = acc[dd] * cs + ps * vr[dd];
      mmax = mn;
    }
    // wave32 merge
#pragma unroll
    for (int off = 16; off >= 1; off >>= 1) {
      const float m2 = __shfl_xor(mmax, off, 32);
      const float s2 = __shfl_xor(ssum, off, 32);
      const float mn = fmaxf(mmax, m2);
      const float f1 = __expf(mmax - mn), f2 = __expf(m2 - mn);
#pragma unroll
      for (int dd = 0; dd < DH_; ++dd) {
        const float a2 = __shfl_xor(acc[dd], off, 32);
        acc[dd] = acc[dd] * f1 + a2 * f2;
      }
      ssum = ssum * f1 + s2 * f2;
      mmax = mn;
    }
    if (lane == 0) {
      const float inv = 1.f / ssum;
#pragma unroll
      for (int dd = 0; dd < DH_; ++dd)
        o[((long)b * C_ + h * DH_ + dd) * 49 + q] = acc[dd] * inv;
    }
  }
}

// ---------------------------------------------------------------------------
// Kernel 6: out = concat(qa, bilinear_up(o)) @ Wproj + bproj. 32 rows/block.
// ---------------------------------------------------------------------------
__global__ __launch_bounds__(256) void k_proj(
    const float* __restrict__ qa, const float* __restrict__ o,
    const float* __restrict__ Wproj, const float* __restrict__ bproj,
    float* __restrict__ out) {
  __shared__ float ct[32 * 260];
  const int tid = threadIdx.x;
  const long row0 = (long)blockIdx.x * 32;
  __builtin_prefetch(Wproj, 0, 3);
  // qa half: 32x128
  for (int i = 0; i < 4; ++i) {
    int idx = i * 256 + tid;
    int r = idx >> 5, c4 = (idx & 31) << 2;
    cp_b128(qa + (row0 + r) * C_ + c4, &ct[r * 260 + c4]);
  }
  // att_up half: half-pixel bilinear from 7x7 grid, edge-clamped taps
  for (int i = 0; i < 16; ++i) {
    const int idx = i * 256 + tid;        // 0..4095
    const int r = idx >> 7, c = idx & 127;
    const long pix = row0 + r;
    const int b = (int)(pix >> 14);
    const int rem = (int)(pix & 16383);
    const int y = rem >> 7, x = rem & 127;
    const float fy = (y + 0.5f) * (7.f / 128.f) - 0.5f;
    const float fx = (x + 0.5f) * (7.f / 128.f) - 0.5f;
    const int y0 = (int)floorf(fy), x0 = (int)floorf(fx);
    const float wy = fy - (float)y0, wx = fx - (float)x0;
    const int y0c = max(y0, 0), y1c = min(y0 + 1, 6);
    const int x0c = max(x0, 0), x1c = min(x0 + 1, 6);
    const float* ob = o + ((long)b * C_ + c) * 49;
    const float v00 = ob[y0c * 7 + x0c], v01 = ob[y0c * 7 + x1c];
    const float v10 = ob[y1c * 7 + x0c], v11 = ob[y1c * 7 + x1c];
    ct[r * 260 + C_ + c] = (1.f - wy) * ((1.f - wx) * v00 + wx * v01) +
                           wy * ((1.f - wx) * v10 + wx * v11);
  }
  cp_wait();
  __syncthreads();
  const int lane = tid & 31, wave = tid >> 5;
  const int half = lane & 15, hi = lane >> 4;
  const int nt = wave;
  v8f acc[2] = {V8F_ZERO, V8F_ZERO};
  wmma_strip<2>(ct, 260, Wproj + nt * 16, C_, 2 * C_, lane, acc);
  const int col = nt * 16 + half;
  const float bias = bproj[col];
#pragma unroll
  for (int mt = 0; mt < 2; ++mt) {
#pragma unroll
    for (int r8 = 0; r8 < 8; ++r8) {
      const long rowg = row0 + mt * 16 + r8 + 8 * hi;
      out[rowg * C_ + col] = acc[mt][r8] + bias;
    }
  }
}

// ---------------------------------------------------------------------------
extern "C" void kernel_launch(void* const* d_in, const int* in_sizes, int n_in,
                              void* d_out, int out_size, void* d_ws,
                              size_t ws_size, hipStream_t stream) {
  const float* x        = (const float*)d_in[0];
  const float* ln_w     = (const float*)d_in[1];
  const float* ln_b     = (const float*)d_in[2];
  const float* Wq       = (const float*)d_in[3];
  const float* bq       = (const float*)d_in[4];
  const float* Wl       = (const float*)d_in[5];
  const float* bl       = (const float*)d_in[6];
  const float* Wa       = (const float*)d_in[7];
  const float* ba       = (const float*)d_in[8];
  const float* conv_w   = (const float*)d_in[9];
  const float* conv_b   = (const float*)d_in[10];
  const float* convatt_w= (const float*)d_in[11];
  const float* convatt_b= (const float*)d_in[12];
  const float* Wkv      = (const float*)d_in[13];
  const float* bkv      = (const float*)d_in[14];
  const float* m        = (const float*)d_in[15];
  const float* Wproj    = (const float*)d_in[16];
  const float* bproj    = (const float*)d_in[17];

  float* ws = (float*)d_ws;
  const size_t NC = (size_t)NPIX * C_;
  float* qbuf  = ws;             // q, then qa (in place)
  float* xlbuf = ws + NC;
  float* apre  = ws + 2 * NC;
  float* bbuf  = ws + 3 * NC;
  float* kvbuf = ws + NC;        // aliases xl+a_pre once both are dead
  float* obuf  = ws + 4 * NC;    // B*C*49 floats
  float* outp  = (float*)d_out;

  k_ln_q_l <<<NPIX / 64, 256, 0, stream>>>(x, ln_w, ln_b, Wq, bq, Wl, bl,
                                           qbuf, xlbuf);
  k_dwconv2<<<(NPIX * 32) / 256, 256, 0, stream>>>(xlbuf, conv_w, conv_b,
                                                   convatt_w, convatt_b,
                                                   apre, bbuf);
  k_gemm_qa<<<NPIX / 64, 256, 0, stream>>>(apre, Wa, ba, qbuf);
  k_gemm_kv<<<NPIX / 64, 256, 0, stream>>>(bbuf, Wkv, bkv, kvbuf);
  k_attn   <<<B_ * NH_, 256, 0, stream>>>(kvbuf, m, obuf);
  k_proj   <<<NPIX / 32, 256, 0, stream>>>(qbuf, obuf, Wproj, bproj, outp);
}